// ScaledDotProductAttention_18528488915140
// MI455X (gfx1250) — compile-verified
//
#include <hip/hip_runtime.h>
#include <hip/hip_bf16.h>
#include <math.h>

// ---------------------------------------------------------------------------
// Causal scaled-dot-product attention, fp32, returning (output, attn).
// B=2, H=16, S=2048, D=64, temperature = 8.0
// Matrix math via V_WMMA_F32_16X16X4_F32 (exact fp32, matches reference).
// attn (537 MB, write-once) stored non-temporally to keep K/V L2-resident.
// ---------------------------------------------------------------------------

typedef __attribute__((ext_vector_type(2))) float v2f;
typedef __attribute__((ext_vector_type(8))) float v8f;

constexpr int kB = 2;
constexpr int kH = 16;
constexpr int kS = 2048;
constexpr int kD = 64;
constexpr int kRows = 16;            // query rows per workgroup
constexpr int kEPad = kS + 4;        // 2052: logits row stride (bank-conflict pad)
constexpr int kQPad = kD + 4;        // 68:   Q-tile row stride
constexpr float kInvTemp = 0.125f;   // 1/8

static __device__ __forceinline__ v8f wmma_f32(v2f a, v2f b, v8f c) {
  // D = A(16x4,f32) * B(4x16,f32) + C(16x16,f32)
  return __builtin_amdgcn_wmma_f32_16x16x4_f32(
      /*neg_a=*/false, a, /*neg_b=*/false, b,
      /*c_mod=*/(short)0, c, /*reuse_a=*/false, /*reuse_b=*/false);
}

__global__ void __launch_bounds__(256)
attn_fwd_kernel(const float* __restrict__ q, const float* __restrict__ kmat,
                const float* __restrict__ vmat, float* __restrict__ out_o,
                float* __restrict__ out_attn, unsigned int* __restrict__ wsmax) {
  extern __shared__ float smem[];
  float* E    = smem;                      // kRows x kEPad  (logits -> exp values)
  float* qred = smem + kRows * kEPad;      // kRows x kQPad  (Q tile, reused as reduce buf)
  float* invs = qred + kRows * kQPad;      // kRows          (1/rowsum)

  const int tid  = threadIdx.x;
  const int lane = tid & 31;
  const int w    = tid >> 5;               // 8 waves
  const int nn   = lane & 15;              // A row / B,D column within tile
  const int hh   = lane >> 4;              // K-pair half select

  const int r0    = blockIdx.x * kRows;    // first query row of this block
  const int bh    = blockIdx.y;            // fused batch*head
  const int ctMax = r0 >> 4;               // last causal column tile (inclusive)
  const int W     = r0 + kRows;            // stored logit width

  const float* Qbh = q    + (size_t)bh * kS * kD;
  const float* Kbh = kmat + (size_t)bh * kS * kD;
  const float* Vbh = vmat + (size_t)bh * kS * kD;

  // ---- stage Q tile into LDS, pre-scaled by 1/temperature -----------------
  for (int idx = tid; idx < kRows * kD; idx += 256) {
    int r = idx >> 6, c = idx & 63;
    qred[r * kQPad + c] = Qbh[(size_t)(r0 + r) * kD + c] * kInvTemp;
  }
  __syncthreads();

  // ---- Phase 1: logits S = (Q/t) K^T, one 16x16 tile per wave iteration ---
  for (int ct = w; ct <= ctMax; ct += 8) {
    const int cb = ct * 16;
    v8f acc = {};
    for (int s2 = 0; s2 < 16; ++s2) {
      const int kk = 4 * s2 + 2 * hh;
      v2f a = *(const v2f*)(qred + nn * kQPad + kk);                 // Q[m][kk..kk+1]
      v2f b = *(const v2f*)(Kbh + (size_t)(cb + nn) * kD + kk);      // K[cb+n][kk..kk+1]
      acc = wmma_f32(a, b, acc);
    }
    for (int j = 0; j < 8; ++j) {
      const int row = j + 8 * hh;          // row within block
      const int col = cb + nn;             // global key column
      float val = acc[j];
      if (col > r0 + row) val = -1e9f;     // causal mask (matches reference)
      E[row * kEPad + col] = val;
    }
  }
  __syncthreads();

  // ---- Phase 2: row softmax (max, exp, sum) over cols [0, W) --------------
  {
    const int i = tid >> 4;                // row 0..15
    const int g = tid & 15;                // 16 threads per row
    float mx = -3.0e38f;
    for (int c = g; c < W; c += 16) mx = fmaxf(mx, E[i * kEPad + c]);
    for (int d = 8; d >= 1; d >>= 1) mx = fmaxf(mx, __shfl_xor(mx, d, 32));
    float sum = 0.0f;
    for (int c = g; c < W; c += 16) {
      float e = expf(E[i * kEPad + c] - mx);
      E[i * kEPad + c] = e;                // keep unnormalized exp in LDS
      sum += e;
    }
    for (int d = 8; d >= 1; d >>= 1) sum += __shfl_xor(sum, d, 32);
    if (g == 0) invs[i] = 1.0f / sum;
  }
  __syncthreads();

  // ---- Phase 2b: write normalized attention block ------------------------
  // Non-temporal: attn is 537 MB write-once data; keep it out of the 192 MB
  // L2 so K/V (33.5 MB) stay resident for every row block's WMMA fetches.
  float* attn_blk = out_attn + ((size_t)bh * kS + r0) * kS;
  for (int idx = tid; idx < kRows * kS; idx += 256) {
    const int r = idx >> 11;
    const int c = idx & (kS - 1);
    const float p = (c < W) ? E[r * kEPad + c] * invs[r] : 0.0f;
    __builtin_nontemporal_store(p, &attn_blk[(size_t)r * kS + c]);
  }

  // ---- Phase 3: O = E * V (then scale rows by 1/sum) ----------------------
  const int nd  = (w & 3) * 16;            // which 16-wide slice of D
  const int par = w >> 2;                  // 2-way K-tile split
  v8f oacc = {};
  for (int ct = par; ct <= ctMax; ct += 2) {
    const int cb = ct * 16;
    for (int s2 = 0; s2 < 16; ++s2) {
      const int kk = 4 * s2 + 2 * hh;
      v2f a = *(const v2f*)(E + nn * kEPad + cb + kk);               // E[m][cb+kk..+1]
      v2f b;
      b.x = Vbh[(size_t)(cb + kk)     * kD + nd + nn];               // V[cb+kk  ][nd+n]
      b.y = Vbh[(size_t)(cb + kk + 1) * kD + nd + nn];               // V[cb+kk+1][nd+n]
      oacc = wmma_f32(a, b, oacc);
    }
  }
  __syncthreads();
  float* red = qred;                       // reuse Q-tile LDS as reduce buffer
  if (w >= 4) {
    const int base = ((w - 4) * 32 + lane) * 8;
    for (int j = 0; j < 8; ++j) red[base + j] = oacc[j];
  }
  __syncthreads();
  if (w < 4) {
    const int base = (w * 32 + lane) * 8;
    float lmax = 0.0f;
    for (int j = 0; j < 8; ++j) {
      const int row = j + 8 * hh;
      const float o = (oacc[j] + red[base + j]) * invs[row];
      out_o[((size_t)bh * kS + r0 + row) * kD + nd + nn] = o;
      lmax = fmaxf(lmax, o);               // clamped at 0: only m>1 matters
    }
    for (int d = 16; d >= 1; d >>= 1) lmax = fmaxf(lmax, __shfl_xor(lmax, d, 32));
    if (lane == 0) atomicMax(wsmax, __float_as_uint(lmax));  // monotone for x>=0
  }
}

__global__ void init_ws_kernel(unsigned int* wsmax) {
  if (threadIdx.x == 0 && blockIdx.x == 0) *wsmax = 0u;
}

__global__ void rescale_kernel(float* __restrict__ out_o,
                               const unsigned int* __restrict__ wsmax, int n) {
  const float m = __uint_as_float(*wsmax);
  const float sc = (m > 1.0f) ? (1.0f / sqrtf(m)) : 1.0f;
  for (int i = blockIdx.x * blockDim.x + threadIdx.x; i < n;
       i += gridDim.x * blockDim.x) {
    const float o = out_o[i] * sc;
    __builtin_nontemporal_store(o, &out_o[i]);  // final write, never re-read
  }
}

extern "C" void kernel_launch(void* const* d_in, const int* in_sizes, int n_in,
                              void* d_out, int out_size, void* d_ws, size_t ws_size,
                              hipStream_t stream) {
  const float* q = (const float*)d_in[0];
  const float* k = (const float*)d_in[1];
  const float* v = (const float*)d_in[2];
  // d_in[3] is the causal mask; causality is applied analytically in-kernel.

  float* out_o    = (float*)d_out;                         // [B,H,S,D]
  float* out_attn = out_o + (size_t)kB * kH * kS * kD;     // [B,H,S,S]
  unsigned int* wsmax = (unsigned int*)d_ws;

  const size_t smem_bytes =
      (size_t)(kRows * kEPad + kRows * kQPad + kRows) * sizeof(float); // ~135.7 KB
  (void)hipFuncSetAttribute((const void*)attn_fwd_kernel,
                            hipFuncAttributeMaxDynamicSharedMemorySize,
                            (int)smem_bytes);

  init_ws_kernel<<<1, 32, 0, stream>>>(wsmax);
  attn_fwd_kernel<<<dim3(kS / kRows, kB * kH), 256, smem_bytes, stream>>>(
      q, k, v, out_o, out_attn, wsmax);
  const int n = kB * kH * kS * kD;
  rescale_kernel<<<dim3(2048), 256, 0, stream>>>(out_o, wsmax, n);
}